// NLGP_80762565034259
// MI455X (gfx1250) — compile-verified
//
#include <hip/hip_runtime.h>
#include <hip/hip_bf16.h>

typedef float v2f __attribute__((ext_vector_type(2)));
typedef float v4f __attribute__((ext_vector_type(4)));
typedef float v8f __attribute__((ext_vector_type(8)));

#define HW     1024
#define C      256
#define B      8
#define NROWS  (B * HW)   // 8192 flattened (batch, pixel) rows
#define NCHUNK 16         // j-chunks of 64 columns each
#define JTILES 4          // 16-wide j tiles per chunk
#define KSLICE 128        // K staged per barrier phase
#define BPAD   132        // 128 + 4 pad floats: row m starts at bank 4m -> conflict-free

// ---------------------------------------------------------------------------
// Kernel 1: g[row, cc] = sum_c x[b,c,p] * w_g[cc,c]   (row = b*HW + p)
// ---------------------------------------------------------------------------
__global__ void k1_proj(const float* __restrict__ x, const float* __restrict__ wg,
                        float* __restrict__ g) {
    int p  = blockIdx.x * 256 + threadIdx.x;  // 0..1023
    int cc = blockIdx.y;                      // 0..255
    int b  = blockIdx.z;                      // 0..7
    const float* xp = x  + (size_t)b * C * HW + p;
    const float* wr = wg + (size_t)cc * C;
    float acc = 0.f;
#pragma unroll 8
    for (int c = 0; c < C; ++c) acc += xp[(size_t)c * HW] * wr[c];
    g[((size_t)b * HW + p) * C + cc] = acc;
}

// ---------------------------------------------------------------------------
// Kernel 2: fused Gram + max-over-b2 + row-sum, fp32 WMMA 16x16x4.
//   * 4 waves / block; each wave owns TWO 16-row A tiles (B reused 2x in reg)
//   * B tiles staged in LDS (one L2 fetch per block, shared by 4 waves),
//     K in 128-wide slices; padded rows (stride 132) -> conflict-free ds_load
//   * A prefetched one K-step ahead (global/L2 latency); B read per step into
//     an 8-entry fragment array (distinct register pairs -> staggered
//     s_wait_dscnt, short LDS latency needs no double buffer)
//   * VGPR budget kept < 256 to avoid s_set_vgpr_msb in the hot loop and
//     keep ~4 waves/SIMD
// ---------------------------------------------------------------------------
__global__ void __launch_bounds__(128) k2_gram(const float* __restrict__ g,
                                               float* __restrict__ partial) {
    __shared__ float Bs[8 * 16 * BPAD];       // 67,584 B

    int tid  = threadIdx.x;                   // 0..127
    int wave = tid >> 5;                      // 0..3
    int lane = tid & 31;
    int m  = lane & 15;                       // row / col within tile
    int kh = lane >> 4;                       // K half-select for 16x16x4
    int chunk = blockIdx.y;                   // 0..15

    int rowBase0 = (blockIdx.x * 8 + wave * 2) * 16;
    int rowBase1 = rowBase0 + 16;
    const float* arow0 = g + (size_t)(rowBase0 + m) * C + 2 * kh;
    const float* arow1 = g + (size_t)(rowBase1 + m) * C + 2 * kh;

    v8f colacc0 = {}, colacc1 = {};

    for (int jt = 0; jt < JTILES; ++jt) {
        int j0 = chunk * 64 + jt * 16;

        v8f acc0[8], acc1[8];
#pragma unroll
        for (int q = 0; q < 8; ++q) { acc0[q] = (v8f){}; acc1[q] = (v8f){}; }

        for (int half = 0; half < 2; ++half) {
            __syncthreads();   // previous slice fully consumed before overwrite
            // ---- cooperative fill: 8 b2-tiles x 16 rows x 128 K floats ----
#pragma unroll 2
            for (int it = 0; it < 32; ++it) {
                int f   = it * 128 + tid;
                int row = f >> 5;              // b2*16 + j
                int k4  = f & 31;
                int b2  = row >> 4;
                int j   = row & 15;
                const v4f* src = (const v4f*)(g + (size_t)(b2 * HW + j0 + j) * C
                                              + half * KSLICE) + k4;
                *((v4f*)(Bs + row * BPAD) + k4) = *src;
            }
            __syncthreads();

            const float* a0p = arow0 + half * KSLICE;
            const float* a1p = arow1 + half * KSLICE;
            const float* bb  = Bs + m * BPAD + 2 * kh;

            v2f a0 = *(const v2f*)(a0p);
            v2f a1 = *(const v2f*)(a1p);

            for (int ks = 0; ks < KSLICE - 4; ks += 4) {
                int kn = ks + 4;
                v2f a0n = *(const v2f*)(a0p + kn);   // prefetch next A step
                v2f a1n = *(const v2f*)(a1p + kn);
                v2f bf[8];                           // 8 distinct reg pairs
#pragma unroll
                for (int q = 0; q < 8; ++q)
                    bf[q] = *(const v2f*)(bb + q * 16 * BPAD + ks);
#pragma unroll
                for (int q = 0; q < 8; ++q) {
                    acc0[q] = __builtin_amdgcn_wmma_f32_16x16x4_f32(
                        false, a0, false, bf[q], (short)0, acc0[q], false, false);
                    acc1[q] = __builtin_amdgcn_wmma_f32_16x16x4_f32(
                        false, a1, false, bf[q], (short)0, acc1[q], false, false);
                }
                a0 = a0n; a1 = a1n;
            }
            {   // epilogue: last K-step of this slice
                v2f bf[8];
#pragma unroll
                for (int q = 0; q < 8; ++q)
                    bf[q] = *(const v2f*)(bb + q * 16 * BPAD + (KSLICE - 4));
#pragma unroll
                for (int q = 0; q < 8; ++q) {
                    acc0[q] = __builtin_amdgcn_wmma_f32_16x16x4_f32(
                        false, a0, false, bf[q], (short)0, acc0[q], false, false);
                    acc1[q] = __builtin_amdgcn_wmma_f32_16x16x4_f32(
                        false, a1, false, bf[q], (short)0, acc1[q], false, false);
                }
            }
        }

        v8f vmax0 = acc0[0], vmax1 = acc1[0];
#pragma unroll
        for (int q = 1; q < 8; ++q) {
#pragma unroll
            for (int e = 0; e < 8; ++e) {
                vmax0[e] = fmaxf(vmax0[e], acc0[q][e]);
                vmax1[e] = fmaxf(vmax1[e], acc1[q][e]);
            }
        }
        colacc0 += vmax0;
        colacc1 += vmax1;
    }

    // D layout: comp e, lanes 0-15 -> row base+e, lanes 16-31 -> base+8+e,
    // column = lane%16. Sum the 16 columns within each half-wave.
    float v0[8], v1[8];
#pragma unroll
    for (int e = 0; e < 8; ++e) {
        float t0 = colacc0[e], t1 = colacc1[e];
        t0 += __shfl_xor(t0, 1, 16);  t1 += __shfl_xor(t1, 1, 16);
        t0 += __shfl_xor(t0, 2, 16);  t1 += __shfl_xor(t1, 2, 16);
        t0 += __shfl_xor(t0, 4, 16);  t1 += __shfl_xor(t1, 4, 16);
        t0 += __shfl_xor(t0, 8, 16);  t1 += __shfl_xor(t1, 8, 16);
        v0[e] = t0; v1[e] = t1;
    }
    if (m == 0) {                      // lanes 0 and 16 of each wave
        int r0 = rowBase0 + kh * 8;
        int r1 = rowBase1 + kh * 8;
#pragma unroll
        for (int e = 0; e < 8; ++e) {
            partial[(size_t)(r0 + e) * NCHUNK + chunk] = v0[e];
            partial[(size_t)(r1 + e) * NCHUNK + chunk] = v1[e];
        }
    }
}

// ---------------------------------------------------------------------------
// Kernel 3: m[k,i] = (scale/b) * sum_chunk partial; softmax over i per batch.
// ---------------------------------------------------------------------------
__global__ void k3_softmax(const float* __restrict__ partial, float* __restrict__ attn) {
    int k = blockIdx.x;    // batch 0..7
    int i = threadIdx.x;   // 0..1023
    __shared__ float red[HW];

    const float* pr = partial + ((size_t)k * HW + i) * NCHUNK;
    float s = 0.f;
#pragma unroll
    for (int e = 0; e < NCHUNK; ++e) s += pr[e];
    float mval = s * 0.0078125f;   // scale/b = (1/16)/8

    red[i] = mval; __syncthreads();
    for (int off = 512; off > 0; off >>= 1) {
        if (i < off) red[i] = fmaxf(red[i], red[i + off]);
        __syncthreads();
    }
    float mx = red[0]; __syncthreads();

    float e = expf(mval - mx);
    red[i] = e; __syncthreads();
    for (int off = 512; off > 0; off >>= 1) {
        if (i < off) red[i] += red[i + off];
        __syncthreads();
    }
    attn[(size_t)k * HW + i] = e / red[0];
}

// ---------------------------------------------------------------------------
// Kernel 4: out[b,d,p] = attn[b,p] * sum_c w_mask[d,c] * x[b,c,p]
// ---------------------------------------------------------------------------
__global__ void k4_out(const float* __restrict__ x, const float* __restrict__ wm,
                       const float* __restrict__ attn, float* __restrict__ out) {
    int p = blockIdx.x * 256 + threadIdx.x;  // 0..1023
    int d = blockIdx.y;                      // 0..255
    int b = blockIdx.z;                      // 0..7
    const float* xp = x  + (size_t)b * C * HW + p;
    const float* wr = wm + (size_t)d * C;
    float acc = 0.f;
#pragma unroll 8
    for (int c = 0; c < C; ++c) acc += xp[(size_t)c * HW] * wr[c];
    out[((size_t)b * C + d) * HW + p] = acc * attn[(size_t)b * HW + p];
}

// ---------------------------------------------------------------------------
extern "C" void kernel_launch(void* const* d_in, const int* in_sizes, int n_in,
                              void* d_out, int out_size, void* d_ws, size_t ws_size,
                              hipStream_t stream) {
    const float* x  = (const float*)d_in[0];
    // d_in[1] = w_phi, d_in[2] = w_theta: provably unused (softmax rows sum
    // to 1, so the mean over M's last axis removes A entirely).
    const float* wg = (const float*)d_in[3];
    const float* wm = (const float*)d_in[4];
    float* out = (float*)d_out;

    float* g       = (float*)d_ws;                       // 8192*256 f32 = 8 MB
    float* partial = g + (size_t)NROWS * C;              // 8192*16 f32 = 512 KB
    float* attn    = partial + (size_t)NROWS * NCHUNK;   // 8192 f32

    k1_proj   <<<dim3(HW / 256, C, B),      256, 0, stream>>>(x, wg, g);
    k2_gram   <<<dim3(NROWS / 128, NCHUNK), 128, 0, stream>>>(g, partial);
    k3_softmax<<<dim3(B),                    HW, 0, stream>>>(partial, attn);
    k4_out    <<<dim3(HW / 256, C, B),      256, 0, stream>>>(x, wm, attn, out);
}